// NeuralNetwork_43705587204567
// MI455X (gfx1250) — compile-verified
//
#include <hip/hip_runtime.h>
#include <hip/hip_bf16.h>
#include <math.h>

#define REFRACTORY 0.33f
#define CLIP_AT    1.0f

#define TPB   256          // 8 waves of 32
#define EPT   4            // edges per thread per tile
#define TILE  (TPB * EPT)  // 1024 edges per tile
#define EDGE_GRID 2048     // persistent blocks

#if __has_builtin(__builtin_amdgcn_tensor_load_to_lds)
#define HAVE_TDM 1
#else
#define HAVE_TDM 0
#endif

typedef unsigned int u32x4 __attribute__((ext_vector_type(4)));
typedef int          i32x8 __attribute__((ext_vector_type(8)));
typedef int          i32x4 __attribute__((ext_vector_type(4)));

// ---------------------------------------------------------------------------
// LDS byte offset of a __shared__ object (generic ptr low 32 bits)
// ---------------------------------------------------------------------------
__device__ __forceinline__ unsigned lds_off(const void* p) {
    return (unsigned)(unsigned long long)p;
}

// ---------------------------------------------------------------------------
// Fallback: per-lane async global->LDS (ASYNCcnt-tracked), known-good asm.
// ---------------------------------------------------------------------------
__device__ __forceinline__ void async_b128(unsigned lds, const void* g) {
    asm volatile("global_load_async_to_lds_b128 %0, %1, off"
                 :: "v"(lds), "v"((unsigned long long)g) : "memory");
}
__device__ __forceinline__ void async_b32(unsigned lds, const void* g) {
    asm volatile("global_load_async_to_lds_b32 %0, %1, off"
                 :: "v"(lds), "v"((unsigned long long)g) : "memory");
}
template <int NCNT>
__device__ __forceinline__ void wait_asynccnt() {
    asm volatile("s_wait_asynccnt %0" :: "i"(NCNT) : "memory");
}

#if HAVE_TDM
// ---------------------------------------------------------------------------
// Tensor Data Mover staging: one instruction moves a whole contiguous stripe
// global->LDS.  D# layout per CDNA5 ISA 8.3/8.4 (normal mode, 1-row 2D tile).
// dsz code: 0=1B, 2=4B elements.  cpol=1 (TH_LOAD_NT): edge streams are
// use-once (272MB/pass) -> keep them out of the 192MB L2 hot set.
// This toolchain's builtin takes 6 args (probe-confirmed arity):
//   (u32x4 grp0, i32x8 grp1, i32x4 grp2, i32x4 grp3, i32x8 extra, i32 cpol)
// grp2/grp3/extra are zero: dims >=2 unused for a 1-row stripe.
// ---------------------------------------------------------------------------
__device__ __forceinline__ void tdm_load(unsigned lds, const void* g,
                                         unsigned elems, int dsz) {
    unsigned long long ga = (unsigned long long)g;
    u32x4 g0;
    g0[0] = 0x1u;                                   // count=1, user mode
    g0[1] = lds;                                    // lds_addr (bytes)
    g0[2] = (unsigned)ga;                           // global_addr[31:0]
    g0[3] = (unsigned)(ga >> 32) | (2u << 30);      // addr[56:32] | type=2
    i32x8 g1;
    g1[0] = dsz << 16;                              // data_size, no multicast
    g1[1] = (int)((elems & 0xFFFFu) << 16);         // tensor_dim0[15:0]
    g1[2] = (int)((elems >> 16) | (1u << 16));      // tensor_dim0[31:16] | tensor_dim1=1
    g1[3] = (int)((elems & 0xFFFFu) << 16);         // tile_dim0
    g1[4] = 1;                                      // tile_dim1=1, tile_dim2=0
    g1[5] = (int)elems;                             // tensor_dim0_stride[31:0]
    g1[6] = 0;
    g1[7] = 0;
    i32x4 z4 = {0, 0, 0, 0};                        // groups 2/3 unused (<=2D)
    i32x8 z8 = {0, 0, 0, 0, 0, 0, 0, 0};
    __builtin_amdgcn_tensor_load_to_lds(g0, g1, z4, z4, z8, /*cpol NT*/1);
}
#endif

// ---------------------------------------------------------------------------
// Kernel 1: acc[i] = 0.33*state[i] + biases[i]
// ---------------------------------------------------------------------------
__global__ void nn_init_kernel(const float* __restrict__ state,
                               const float* __restrict__ biases,
                               float* __restrict__ acc, int n) {
    int i = (blockIdx.x * blockDim.x + threadIdx.x) * 4;
    if (i + 3 < n) {
        float4 s = *(const float4*)(state + i);
        float4 b = *(const float4*)(biases + i);
        float4 r;
        r.x = fmaf(REFRACTORY, s.x, b.x);
        r.y = fmaf(REFRACTORY, s.y, b.y);
        r.z = fmaf(REFRACTORY, s.z, b.z);
        r.w = fmaf(REFRACTORY, s.w, b.w);
        *(float4*)(acc + i) = r;
    } else {
        for (; i < n; ++i) acc[i] = fmaf(REFRACTORY, state[i], biases[i]);
    }
}

// ---------------------------------------------------------------------------
// Kernel 2: edge scatter with double-buffered TDM (or async-LDS) staging.
// ---------------------------------------------------------------------------
__global__ void __launch_bounds__(TPB)
nn_edge_kernel(const float* __restrict__ state,
               const float* __restrict__ W,
               const float* __restrict__ inp,
               const int*   __restrict__ S,
               const int*   __restrict__ D,
               const unsigned char* __restrict__ M,
               const int*   __restrict__ I,
               float* __restrict__ acc,
               long long nfull_tiles, long long E) {
    __shared__ float         sw[2][TILE];
    __shared__ int           ss[2][TILE];
    __shared__ int           sd[2][TILE];
    __shared__ int           si[2][TILE];
    __shared__ unsigned char sm[2][TILE];

    const int tid  = threadIdx.x;
    const int slot = tid * EPT;

#if HAVE_TDM
    const bool issuer = (tid >> 5) == 0;   // wave 0 drives the TDM
    auto stage = [&](int b, long long t) {
        long long e = t * TILE;
        tdm_load(lds_off(&sw[b][0]), W + e, TILE, 2);
        tdm_load(lds_off(&ss[b][0]), S + e, TILE, 2);
        tdm_load(lds_off(&sd[b][0]), D + e, TILE, 2);
        tdm_load(lds_off(&si[b][0]), I + e, TILE, 2);
        tdm_load(lds_off(&sm[b][0]), M + e, TILE, 0);
    };
#else
    auto stage = [&](int b, long long t) {
        long long e = t * TILE + (long long)slot;
        async_b128(lds_off(&sw[b][slot]), W + e);
        async_b128(lds_off(&ss[b][slot]), S + e);
        async_b128(lds_off(&sd[b][slot]), D + e);
        async_b128(lds_off(&si[b][slot]), I + e);
        async_b32 (lds_off(&sm[b][slot]), M + e);
    };
#endif

    long long tile = blockIdx.x;
    const long long tstride = gridDim.x;
    int buf = 0;
    bool have = (tile < nfull_tiles);

#if HAVE_TDM
    if (issuer && have) stage(0, tile);
#else
    if (have) stage(0, tile);
#endif

    while (have) {
        long long next = tile + tstride;
        bool have_next = (next < nfull_tiles);
#if HAVE_TDM
        if (issuer) {
            if (have_next) {
                stage(buf ^ 1, next);
                __builtin_amdgcn_s_wait_tensorcnt(5);  // current buf complete
            } else {
                __builtin_amdgcn_s_wait_tensorcnt(0);
            }
        }
        __syncthreads();                               // publish buf to all waves
#else
        if (have_next) {
            stage(buf ^ 1, next);
            wait_asynccnt<5>();
        } else {
            wait_asynccnt<0>();
        }
#endif
#pragma unroll
        for (int k = 0; k < EPT; ++k) {
            int j = slot + k;
            float w = sw[buf][j];
            int   s = ss[buf][j];
            int   d = sd[buf][j];
            int  ii = si[buf][j];
            float from = sm[buf][j] ? (REFRACTORY * state[s]) : inp[ii];
            __hip_atomic_fetch_add(&acc[d], w * from,
                                   __ATOMIC_RELAXED, __HIP_MEMORY_SCOPE_AGENT);
        }
#if HAVE_TDM
        __syncthreads();    // all waves done with buf before TDM rewrites it
#endif
        tile = next;
        buf ^= 1;
        have = have_next;
    }

    // Tail edges (E % TILE), direct loads (E=16M is tile-exact; kept general)
    long long tail0 = nfull_tiles * TILE;
    for (long long e = tail0 + (long long)blockIdx.x * TPB + tid; e < E;
         e += (long long)gridDim.x * TPB) {
        float w = W[e];
        float from = M[e] ? (REFRACTORY * state[S[e]]) : inp[I[e]];
        __hip_atomic_fetch_add(&acc[D[e]], w * from,
                               __ATOMIC_RELAXED, __HIP_MEMORY_SCOPE_AGENT);
    }
}

// ---------------------------------------------------------------------------
// Kernel 3: activation at the gathered neurons only (8192 of 1M)
// ---------------------------------------------------------------------------
__global__ void nn_out_kernel(const float* __restrict__ acc,
                              const int* __restrict__ act_id,
                              const int* __restrict__ out_ids,
                              float* __restrict__ out, int nout) {
    int j = blockIdx.x * blockDim.x + threadIdx.x;
    if (j >= nout) return;
    int id = out_ids[j];
    float x = acc[id];
    float r;
    switch (act_id[id]) {
        case 0: r = x; break;
        case 1: r = fmaxf(x, 0.0f); break;
        case 2: r = (x > 0.0f) ? x : 0.01f * x; break;                 // leaky_relu
        case 3: r = fminf(fmaxf(x, 0.0f), CLIP_AT); break;             // clipped_relu
        case 4: r = tanhf(x); break;
        case 5: r = 1.0f / (1.0f + __expf(-x)); break;                 // sigmoid
        case 6: r = fmaxf(x, 0.0f) + log1pf(__expf(-fabsf(x))); break; // softplus
        default: r = fabsf(x); break;                                  // abs
    }
    out[j] = r;
}

// ---------------------------------------------------------------------------
extern "C" void kernel_launch(void* const* d_in, const int* in_sizes, int n_in,
                              void* d_out, int out_size, void* d_ws, size_t ws_size,
                              hipStream_t stream) {
    const float*         state   = (const float*)d_in[0];
    const float*         weights = (const float*)d_in[1];
    const float*         biases  = (const float*)d_in[2];
    const float*         inp     = (const float*)d_in[3];
    const int*           src     = (const int*)d_in[4];
    const int*           dst     = (const int*)d_in[5];
    const unsigned char* rec     = (const unsigned char*)d_in[6]; // JAX bool = 1 byte
    const int*           iidx    = (const int*)d_in[7];
    const int*           act     = (const int*)d_in[8];
    const int*           oids    = (const int*)d_in[9];
    float*               out     = (float*)d_out;
    float*               acc     = (float*)d_ws;  // N floats of scratch

    const int       n = in_sizes[0];
    const long long E = (long long)in_sizes[1];

    int initGrid = (n / 4 + TPB - 1) / TPB + 1;
    nn_init_kernel<<<initGrid, TPB, 0, stream>>>(state, biases, acc, n);

    long long nfull = E / TILE;
    nn_edge_kernel<<<EDGE_GRID, TPB, 0, stream>>>(state, weights, inp, src, dst,
                                                  rec, iidx, acc, nfull, E);

    nn_out_kernel<<<(out_size + TPB - 1) / TPB, TPB, 0, stream>>>(acc, act, oids,
                                                                  out, out_size);
}